// ECN_37391985279547
// MI455X (gfx1250) — compile-verified
//
#include <hip/hip_runtime.h>
#include <math.h>

typedef __attribute__((ext_vector_type(2))) float v2f;
typedef __attribute__((ext_vector_type(8))) float v8f;

#define GN     128   // nodes per graph
#define FIN    59    // input features
#define KPAD   64    // padded K for WMMA (16 steps of 4)
#define XPITCH 68    // LDS pitch for xS (avoids bank conflicts on A-frag loads)
#define C1     128   // layer-1 width
#define NCOLS  256   // A|B concatenated GEMM output width
#define BNEPS  1e-5f

// ---------------------------------------------------------------------------
// Prep: build transposed concat weight WcatT[256][64]:
//   cols c<128  : (w1a - w1b)[k][c]   (A-path:  xi @ (W1a - W1b))
//   cols c>=128 : w1b[k][c-128]       (B-path:  xj @ W1b)
// zero-padded for k in [59,64). Stored transposed: WcatT[c*64 + k].
// ---------------------------------------------------------------------------
__global__ __launch_bounds__(256) void ecn_wcat_prep(const float* __restrict__ w1,
                                                     float* __restrict__ wcatT) {
  int t = blockIdx.x * 256 + threadIdx.x;     // 64 blocks * 256 = 64*256 entries
  int k = t >> 8;
  int c = t & 255;
  float v = 0.f;
  if (k < FIN) {
    if (c < C1) v = w1[k * C1 + c] - w1[(k + FIN) * C1 + c];
    else        v = w1[(k + FIN) * C1 + (c - C1)];
  }
  wcatT[c * KPAD + k] = v;
}

// ---------------------------------------------------------------------------
// Fused per-graph kernel: KNN -> WMMA GEMM (A|B) -> edge relu/BN/mean
//   -> global mean pool -> MLP layer2 -> layer3 -> sigmoid
// 256 threads = 8 wave32 per block, one block per graph.
// ---------------------------------------------------------------------------
__global__ __launch_bounds__(256) void ecn_fused(
    const float* __restrict__ x,   const float* __restrict__ pos,
    const float* __restrict__ wcatT,
    const float* __restrict__ b1,  const float* __restrict__ g1,
    const float* __restrict__ be1, const float* __restrict__ m1, const float* __restrict__ v1,
    const float* __restrict__ w2,  const float* __restrict__ b2, const float* __restrict__ g2,
    const float* __restrict__ be2, const float* __restrict__ m2, const float* __restrict__ v2,
    const float* __restrict__ w3,  const float* __restrict__ b3, const float* __restrict__ g3,
    const float* __restrict__ be3, const float* __restrict__ m3, const float* __restrict__ v3,
    float* __restrict__ out)
{
  __shared__ float xS[GN][XPITCH];     //  34816 B : node features, K-padded
  __shared__ float ABs[GN][NCOLS];     // 131072 B : [A | B] GEMM result
  __shared__ float posS[GN * 3];
  __shared__ float sqS[GN];
  __shared__ int   idxS[GN * 3];
  __shared__ float x2S[C1];
  __shared__ float redS[256];

  const int g    = blockIdx.x;
  const int tid  = threadIdx.x;
  const int lane = tid & 31;
  const int wave = tid >> 5;

  // ---- stage inputs into LDS ----
  for (int t = tid; t < GN * 3; t += 256) posS[t] = pos[(size_t)g * GN * 3 + t];
  for (int t = tid; t < GN * XPITCH; t += 256) {
    int i = t / XPITCH, f = t - i * XPITCH;
    xS[i][f] = (f < FIN) ? x[((size_t)g * GN + i) * FIN + f] : 0.f;
  }
  __syncthreads();

  // ---- KNN (k=3, loop=False), matching reference d2 = |pi|^2+|pj|^2-2<pi,pj> ----
  if (tid < GN) {
    float px = posS[tid * 3], py = posS[tid * 3 + 1], pz = posS[tid * 3 + 2];
    sqS[tid] = px * px + py * py + pz * pz;
  }
  __syncthreads();
  if (tid < GN) {
    float px = posS[tid * 3], py = posS[tid * 3 + 1], pz = posS[tid * 3 + 2];
    float sqi = sqS[tid];
    float bd0 = 1e38f, bd1 = 1e38f, bd2 = 1e38f;
    int   bi0 = 0,     bi1 = 0,     bi2 = 0;
    for (int j = 0; j < GN; ++j) {
      if (j == tid) continue;
      float dot = px * posS[j * 3] + py * posS[j * 3 + 1] + pz * posS[j * 3 + 2];
      float d2  = sqi + sqS[j] - 2.f * dot;
      if (d2 < bd0)      { bd2 = bd1; bi2 = bi1; bd1 = bd0; bi1 = bi0; bd0 = d2; bi0 = j; }
      else if (d2 < bd1) { bd2 = bd1; bi2 = bi1; bd1 = d2;  bi1 = j; }
      else if (d2 < bd2) { bd2 = d2;  bi2 = j; }
    }
    idxS[tid * 3 + 0] = bi0;
    idxS[tid * 3 + 1] = bi1;
    idxS[tid * 3 + 2] = bi2;
  }
  __syncthreads();

  // ---- WMMA GEMM: ABs[128][256] = xS[128][64] @ Wcat[64][256] (fp32, 16x16x4) ----
  // A frag: lane%16 = M row, lane/16 selects K pair {k,k+1} vs {k+2,k+3}
  // B frag: lane%16 = N col, same K-pair selection
  // D:      c[r] -> (M = r + 8*(lane/16), N = lane%16)
  {
    const int mrow  = wave * 16 + (lane & 15);
    const int khalf = (lane >> 4) * 2;
    for (int nt = 0; nt < NCOLS / 16; ++nt) {
      const int col = nt * 16 + (lane & 15);
      const float* __restrict__ wcol = wcatT + (size_t)col * KPAD;
      v8f acc = {};
#pragma unroll
      for (int kk = 0; kk < KPAD / 4; ++kk) {
        const int kOff = kk * 4 + khalf;
        v2f a = *(const v2f*)(&xS[mrow][kOff]);
        v2f b = *(const v2f*)(wcol + kOff);
        acc = __builtin_amdgcn_wmma_f32_16x16x4_f32(
            /*neg_a=*/false, a, /*neg_b=*/false, b,
            /*c_mod=*/(short)0, acc, /*reuse_a=*/false, /*reuse_b=*/false);
      }
      const int rbase = wave * 16 + 8 * (lane >> 4);
      const int cw    = nt * 16 + (lane & 15);
#pragma unroll
      for (int r = 0; r < 8; ++r) ABs[rbase + r][cw] = acc[r];
    }
  }
  __syncthreads();

  // ---- edge MLP (pre-act = A_i + B_j + b1), relu, BN+mean folded ----
  // x2[c] = s1[c] * (sum_{i,k} relu(pre)) / (3*128) + t1[c]
  {
    const int c    = tid & 127;
    const int half = tid >> 7;           // split nodes between thread halves
    const float bc = b1[c];
    float accE = 0.f;
    for (int i = half; i < GN; i += 2) {
      float Ai = ABs[i][c] + bc;
      int j0 = idxS[i * 3 + 0], j1 = idxS[i * 3 + 1], j2 = idxS[i * 3 + 2];
      accE += fmaxf(Ai + ABs[j0][C1 + c], 0.f);
      accE += fmaxf(Ai + ABs[j1][C1 + c], 0.f);
      accE += fmaxf(Ai + ABs[j2][C1 + c], 0.f);
    }
    redS[tid] = accE;
  }
  __syncthreads();
  if (tid < C1) {
    float tot = redS[tid] + redS[tid + 128];
    float s1  = g1[tid] * rsqrtf(v1[tid] + BNEPS);
    float t1  = be1[tid] - m1[tid] * s1;
    x2S[tid]  = s1 * (tot * (1.f / (3.f * (float)GN))) + t1;
  }
  __syncthreads();

  // ---- layer 2: h2 = BN(relu(x2 @ w2 + b2)), one output channel per thread ----
  float h2v;
  {
    const int d = tid;                    // 256 output channels
    float acc = b2[d];
#pragma unroll 4
    for (int c0 = 0; c0 < C1; ++c0) acc += x2S[c0] * w2[c0 * 256 + d];
    float r  = fmaxf(acc, 0.f);
    float s2 = g2[d] * rsqrtf(v2[d] + BNEPS);
    float t2 = be2[d] - m2[d] * s2;
    h2v = s2 * r + t2;
  }

  // ---- layer 3: scalar reduce, BN, sigmoid ----
  redS[tid] = h2v * w3[tid];
  __syncthreads();
  for (int s = 128; s > 0; s >>= 1) {
    if (tid < s) redS[tid] += redS[tid + s];
    __syncthreads();
  }
  if (tid == 0) {
    float v  = redS[0] + b3[0];
    float r  = fmaxf(v, 0.f);
    float s3 = g3[0] * rsqrtf(v3[0] + BNEPS);
    float t3 = be3[0] - m3[0] * s3;
    float z  = s3 * r + t3;
    out[g] = 1.f / (1.f + expf(-z));
  }
}

// ---------------------------------------------------------------------------
extern "C" void kernel_launch(void* const* d_in, const int* in_sizes, int n_in,
                              void* d_out, int out_size, void* d_ws, size_t ws_size,
                              hipStream_t stream) {
  const float* x_p   = (const float*)d_in[0];
  const float* pos_p = (const float*)d_in[1];
  const float* w1_p  = (const float*)d_in[2];
  const float* b1_p  = (const float*)d_in[3];
  const float* g1_p  = (const float*)d_in[4];
  const float* be1_p = (const float*)d_in[5];
  const float* m1_p  = (const float*)d_in[6];
  const float* v1_p  = (const float*)d_in[7];
  const float* w2_p  = (const float*)d_in[8];
  const float* b2_p  = (const float*)d_in[9];
  const float* g2_p  = (const float*)d_in[10];
  const float* be2_p = (const float*)d_in[11];
  const float* m2_p  = (const float*)d_in[12];
  const float* v2_p  = (const float*)d_in[13];
  const float* w3_p  = (const float*)d_in[14];
  const float* b3_p  = (const float*)d_in[15];
  const float* g3_p  = (const float*)d_in[16];
  const float* be3_p = (const float*)d_in[17];
  const float* m3_p  = (const float*)d_in[18];
  const float* v3_p  = (const float*)d_in[19];
  float* out = (float*)d_out;

  const int G = in_sizes[1] / (3 * GN);   // pos has G*n*3 elements, n == 128

  float* wcatT = (float*)d_ws;            // 256*64*4 = 64 KB scratch

  ecn_wcat_prep<<<(NCOLS * KPAD) / 256, 256, 0, stream>>>(w1_p, wcatT);
  ecn_fused<<<G, 256, 0, stream>>>(
      x_p, pos_p, wcatT,
      b1_p, g1_p, be1_p, m1_p, v1_p,
      w2_p, b2_p, g2_p, be2_p, m2_p, v2_p,
      w3_p, b3_p, g3_p, be3_p, m3_p, v3_p,
      out);
}